// LinearRegressionHospi_4801773437620
// MI455X (gfx1250) — compile-verified
//
#include <hip/hip_runtime.h>

typedef __attribute__((ext_vector_type(2))) float v2f;
typedef __attribute__((ext_vector_type(8))) float v8f;

namespace {
constexpr int   W       = 30;     // window_size (fixed by reference setup)
constexpr int   NF      = 14;     // n_forecast
constexpr int   T       = 60;     // time steps per series
constexpr int   F       = 4;      // features per step
constexpr int   TIDX    = 0;      // target feature index
constexpr float XMEAN   = 14.5f;            // mean of arange(30)
constexpr float INV_SXX = 1.0f / 2247.5f;   // 1 / sum((k-14.5)^2, k=0..29)
}

// One wave (32 lanes) computes a 16-row tile of result = Y(16x30) * M(30x14)
// using 8 chained V_WMMA_F32_16X16X4_F32 (K padded to 32, N padded to 16).
// M[k][n] = 1/W + (k - XMEAN)*(n + W - XMEAN)/SXX is generated in registers.
__global__ __launch_bounds__(256) void linreg_forecast_wmma(
    const float* __restrict__ in, float* __restrict__ out,
    int numTiles, long long B)
{
    const int lane = threadIdx.x & 31;
    const int wave = threadIdx.x >> 5;
    const int tile = blockIdx.x * 8 + wave;
    if (tile >= numTiles) return;            // wave-uniform exit

    const int mrow = lane & 15;              // A row / D column index
    const int half = lane >> 4;              // 0: lanes 0-15, 1: lanes 16-31

    // Batch row this lane loads for the A fragment (clamped for tail tiles;
    // branch-free, loads stay in bounds, stores are predicated separately).
    long long batch = (long long)tile * 16 + mrow;
    if (batch > B - 1) batch = B - 1;

    // y[b][k] = in[b*T*F + (T-W+k)*F + TIDX]
    const float* base = in + batch * (long long)(T * F)
                           + (long long)(T - W) * F + TIDX;

    // Per-lane B-matrix generators (N column shares lane&15 with A's row).
    // M[k][n] = bc + k*s, with s,bc pre-masked to 0 for padded columns n>=NF.
    const int   n     = mrow;
    const bool  nok   = (n < NF);
    const float s_raw = ((float)n + (float)W - XMEAN) * INV_SXX;
    const float s     = nok ? s_raw : 0.0f;
    const float bc    = nok ? (1.0f / (float)W - XMEAN * s_raw) : 0.0f;

    v8f acc = (v8f)0.0f;

    // bx = M[2*half][n]; advances by 4*s per K-chunk.
    float bx = bc + (float)(2 * half) * s;

    #pragma unroll
    for (int c = 0; c < 8; ++c) {
        const int kb = 4 * c + 2 * half;     // K of A/B element 0
        // A fragment loads; chunk 7 upper half (k=30,31) clamps the address
        // (the value is killed by B==0 there).
        int k0 = kb, k1 = kb + 1;
        if (c == 7) { k0 = half ? 0 : k0; k1 = half ? 0 : k1; }
        v2f a;
        a.x = base[k0 * F];
        a.y = base[k1 * F];
        // B fragment: two adds per chunk; mask padded K only where it exists.
        float b0 = bx;
        float b1 = bx + s;
        if (c == 7) { b0 = half ? 0.0f : b0; b1 = half ? 0.0f : b1; }
        v2f b;
        b.x = b0;
        b.y = b1;
        // D = A x B + C  (f32, 16x16x4)
        acc = __builtin_amdgcn_wmma_f32_16x16x4_f32(
            /*neg_a=*/false, a, /*neg_b=*/false, b,
            /*c_mod=*/(short)0, acc, /*reuse_a=*/false, /*reuse_b=*/false);
        bx += 4.0f * s;
    }

    // D layout: VGPR v -> row (v + 8*half), col n (lanes 0-15 / 16-31 mirror).
    const long long row0 = (long long)tile * 16 + 8 * half;
    float* op = out + row0 * NF + n;

    // Wave-uniform: all 16 rows of this tile valid? (Always true when B%16==0,
    // e.g. B = 1,000,000 -> every wave takes the clause-of-stores fast path.)
    const bool fullTile = ((long long)(tile + 1) * 16) <= B;
    if (fullTile) {
        if (nok) {                       // single EXEC-mask set, then 8 stores
            #pragma unroll
            for (int v = 0; v < 8; ++v)
                op[v * NF] = acc[v];     // immediate offsets 0,56,...,392 B
        }
    } else {                             // at most one tile in the whole grid
        if (nok) {
            #pragma unroll
            for (int v = 0; v < 8; ++v)
                if (row0 + v < B) op[v * NF] = acc[v];
        }
    }
}

extern "C" void kernel_launch(void* const* d_in, const int* in_sizes, int n_in,
                              void* d_out, int out_size, void* d_ws, size_t ws_size,
                              hipStream_t stream) {
    const float* in  = (const float*)d_in[0];
    float*       out = (float*)d_out;

    const long long B        = (long long)in_sizes[0] / (long long)(T * F); // 1,000,000
    const int       numTiles = (int)((B + 15) / 16);                        // 62,500
    const int       blocks   = (numTiles + 7) / 8;                          // 8 waves/block

    linreg_forecast_wmma<<<blocks, 256, 0, stream>>>(in, out, numTiles, B);
}